// StochasticNetworkLoading_83562883711882
// MI455X (gfx1250) — compile-verified
//
#include <hip/hip_runtime.h>

// ---------------------------------------------------------------------------
// StochasticNetworkLoading on MI455X (gfx1250, wave32, WMMA)
//
//   V = (I-L)^-1 via Neumann product form:  P <- P + P*B ;  B <- B*B
//   (6 doubling steps cover sum_{k<=127} L^k; ||L||inf < 0.8 so error ~2e-13)
//   T = Q .* (V==0 ? 1 : 1/V) ;  A = V^T T ;  W = A V^T ;  out = L .* W
//
// All matmuls use the native fp32 matrix op V_WMMA_F32_16X16X4_F32 to match
// the fp32 reference precision. 128x128 block tiles, 8 waves, 2x4 WMMA tiles
// per wave. A staged in padded rows (K-pairs adjacent -> ds_load_2addr into
// operand pair); B staged pair-interleaved by K (one aligned ds_load_b64 per
// fragment, no repack movs). Non-transposed global loads are b128-vectorized.
// ---------------------------------------------------------------------------

#define NDIM   1024
#define NSAMP  16
#define NTRIPS 50000

typedef float v2f __attribute__((ext_vector_type(2)));
typedef float v4f __attribute__((ext_vector_type(4)));
typedef float v8f __attribute__((ext_vector_type(8)));

// -------------------------- elementwise kernels ----------------------------

__global__ void zero_f32(float* __restrict__ p, unsigned n) {
  unsigned i = blockIdx.x * blockDim.x + threadIdx.x;
  if (i < n) p[i] = 0.0f;
}

__global__ void scatter_q(const float* __restrict__ dsq,
                          const int* __restrict__ trips,
                          float* __restrict__ Q, int ntrips) {
  int i = blockIdx.x * blockDim.x + threadIdx.x;
  if (i < ntrips) {
    float d = dsq[i];
    int r = trips[2 * i + 0];
    int s = trips[2 * i + 1];
    atomicAdd(&Q[(size_t)r * NDIM + s], d * d);
  }
}

// P = L + I (per sample), B = L
__global__ void init_PB(const float* __restrict__ L,
                        float* __restrict__ P, float* __restrict__ B) {
  unsigned i = blockIdx.x * blockDim.x + threadIdx.x; // < NSAMP*NDIM*NDIM
  unsigned rc = i & (NDIM * NDIM - 1);
  unsigned r = rc >> 10, c = rc & (NDIM - 1);
  float l = L[i];
  B[i] = l;
  P[i] = l + ((r == c) ? 1.0f : 0.0f);
}

// T = Q .* (V==0 ? 1 : 1/V)   (Q broadcast over samples)
__global__ void make_T(const float* __restrict__ Q,
                       const float* __restrict__ V, float* __restrict__ T) {
  unsigned i = blockIdx.x * blockDim.x + threadIdx.x;
  unsigned rc = i & (NDIM * NDIM - 1);
  float v = V[i];
  float norm = (v == 0.0f) ? 1.0f : (1.0f / v);
  T[i] = Q[rc] * norm;
}

// out = L .* W
__global__ void final_mul(const float* __restrict__ L,
                          const float* __restrict__ W, float* __restrict__ out) {
  unsigned i = blockIdx.x * blockDim.x + threadIdx.x;
  out[i] = L[i] * W[i];
}

// ------------------------------ WMMA GEMM ----------------------------------
// D = opA(A) * opB(B) [+ Cin],  batched over blockIdx.z, all matrices 1024x1024.
// Block: 128x128 C-tile, 256 threads = 8 wave32s (4x2), each wave 32x64 =
// 2x4 V_WMMA_F32_16X16X4_F32 tiles. K staged via LDS in chunks of 16 with a
// register-staged pipeline (load chunk k+1 to regs while computing chunk k).

template <bool TA, bool TB, bool ADD>
__global__ __launch_bounds__(256)
void gemm128(const float* __restrict__ Ag, const float* __restrict__ Bg,
             const float* __restrict__ Cg, float* __restrict__ Dg) {
  const size_t zo = (size_t)blockIdx.z * NDIM * NDIM;
  Ag += zo; Bg += zo; Dg += zo;
  if (ADD) Cg += zo;

  const int mTile = blockIdx.y * 128;
  const int nTile = blockIdx.x * 128;

  __shared__ float As[128][17];      // padded rows: K-pair adjacent, cols conflict-free
  __shared__ float Bs[8][128][2];    // pair-interleaved by K: (k,k+1)@n is one b64

  const int tid  = threadIdx.x;   // 0..255
  const int lane = tid & 31;      // wave32
  const int wave = tid >> 5;      // 0..7
  const int wm   = (wave >> 1) * 32;  // wave row offset: 0/32/64/96
  const int wn   = (wave & 1) * 64;   // wave col offset: 0/64
  const int kh   = lane >> 4;     // lane half: selects K pair in A/B frags
  const int l16  = lane & 15;

  // Staging: each thread owns 4 consecutive elements x 2 groups for A and B.
  const int sm  = tid >> 2;        // A row (group g adds 64):  0..63
  const int sk  = (tid & 3) * 4;   // A col base:               0/4/8/12
  const int sbk = tid >> 5;        // B row (group g adds 8):   0..7
  const int sbn = (tid & 31) * 4;  // B col base:               0..124

  v8f acc[2][4];
#pragma unroll
  for (int i = 0; i < 2; ++i)
#pragma unroll
    for (int j = 0; j < 4; ++j) {
#pragma unroll
      for (int r = 0; r < 8; ++r) {
        if (ADD) {
          // C/D layout: VGPR r holds M = r (lanes 0-15) / M = 8+r (lanes 16-31)
          int m = mTile + wm + i * 16 + kh * 8 + r;
          int n = nTile + wn + j * 16 + l16;
          acc[i][j][r] = Cg[(size_t)m * NDIM + n];
        } else {
          acc[i][j][r] = 0.0f;
        }
      }
    }

  v4f ra[2], rb[2];
  // Prologue: stage chunk 0 into registers
#pragma unroll
  for (int g = 0; g < 2; ++g) {
    const int m = sm + g * 64, kb = sbk + g * 8;
    if (TA) {
#pragma unroll
      for (int j = 0; j < 4; ++j)
        ra[g][j] = Ag[(size_t)(sk + j) * NDIM + (mTile + m)];
    } else {
      ra[g] = *(const v4f*)&Ag[(size_t)(mTile + m) * NDIM + sk];
    }
    if (TB) {
#pragma unroll
      for (int j = 0; j < 4; ++j)
        rb[g][j] = Bg[(size_t)(nTile + sbn + j) * NDIM + kb];
    } else {
      rb[g] = *(const v4f*)&Bg[(size_t)kb * NDIM + (nTile + sbn)];
    }
  }

  for (int k0 = 0; k0 < NDIM; k0 += 16) {
    // Commit staged registers to LDS
    __syncthreads();
#pragma unroll
    for (int g = 0; g < 2; ++g) {
      const int m = sm + g * 64, kb = sbk + g * 8;
#pragma unroll
      for (int j = 0; j < 4; ++j) {
        As[m][sk + j] = ra[g][j];
        Bs[kb >> 1][sbn + j][kb & 1] = rb[g][j];
      }
    }
    __syncthreads();

    // Prefetch chunk k0+32 cachelines (global_prefetch_b8)
    if (k0 + 32 < NDIM) {
      const int kp = k0 + 32;
      __builtin_prefetch(TA ? &Ag[(size_t)(kp + sk) * NDIM + (mTile + sm)]
                            : &Ag[(size_t)(mTile + sm) * NDIM + kp + sk], 0, 3);
      __builtin_prefetch(TB ? &Bg[(size_t)(nTile + sbn) * NDIM + kp + sbk]
                            : &Bg[(size_t)(kp + sbk) * NDIM + (nTile + sbn)], 0, 3);
    }

    // Stage chunk k0+16 into registers while computing chunk k0 from LDS
    if (k0 + 16 < NDIM) {
      const int kn = k0 + 16;
#pragma unroll
      for (int g = 0; g < 2; ++g) {
        const int m = sm + g * 64, kb = sbk + g * 8;
        if (TA) {
#pragma unroll
          for (int j = 0; j < 4; ++j)
            ra[g][j] = Ag[(size_t)(kn + sk + j) * NDIM + (mTile + m)];
        } else {
          ra[g] = *(const v4f*)&Ag[(size_t)(mTile + m) * NDIM + kn + sk];
        }
        if (TB) {
#pragma unroll
          for (int j = 0; j < 4; ++j)
            rb[g][j] = Bg[(size_t)(nTile + sbn + j) * NDIM + kn + kb];
        } else {
          rb[g] = *(const v4f*)&Bg[(size_t)(kn + kb) * NDIM + (nTile + sbn)];
        }
      }
    }

    // Compute: 4 k-steps x (2x4) WMMA tiles = 32 WMMAs per chunk per wave
#pragma unroll
    for (int kk = 0; kk < 16; kk += 4) {
      const int kfrag = kk + 2 * kh;   // lanes 0-15 -> K=kk+0/1, 16-31 -> kk+2/3
      v2f a[2], b[4];
#pragma unroll
      for (int i = 0; i < 2; ++i) {
        int m = wm + i * 16 + l16;
        a[i].x = As[m][kfrag + 0];     // adjacent -> ds_load_2addr pair
        a[i].y = As[m][kfrag + 1];
      }
#pragma unroll
      for (int j = 0; j < 4; ++j) {
        int n = wn + j * 16 + l16;
        b[j] = *(const v2f*)&Bs[kfrag >> 1][n][0];   // one aligned ds_load_b64
      }
#pragma unroll
      for (int i = 0; i < 2; ++i)
#pragma unroll
        for (int j = 0; j < 4; ++j)
          acc[i][j] = __builtin_amdgcn_wmma_f32_16x16x4_f32(
              /*neg_a=*/false, a[i], /*neg_b=*/false, b[j],
              /*c_mod=*/(short)0, acc[i][j],
              /*reuse_a=*/false, /*reuse_b=*/false);
    }
  }

#pragma unroll
  for (int i = 0; i < 2; ++i)
#pragma unroll
    for (int j = 0; j < 4; ++j)
#pragma unroll
      for (int r = 0; r < 8; ++r) {
        int m = mTile + wm + i * 16 + kh * 8 + r;
        int n = nTile + wn + j * 16 + l16;
        Dg[(size_t)m * NDIM + n] = acc[i][j][r];
      }
}

// ------------------------------ host driver --------------------------------

extern "C" void kernel_launch(void* const* d_in, const int* in_sizes, int n_in,
                              void* d_out, int out_size, void* d_ws, size_t ws_size,
                              hipStream_t stream) {
  (void)in_sizes; (void)n_in; (void)out_size; (void)ws_size;

  const float* L    = (const float*)d_in[0];  // [16,1024,1024] f32
  const float* dsq  = (const float*)d_in[1];  // [50000] f32
  const int*   trip = (const int*)d_in[2];    // [50000,2] i32
  float*       out  = (float*)d_out;          // [16,1024,1024] f32

  const size_t NN  = (size_t)NDIM * NDIM;     // 1M elems, 4 MB
  const size_t BNN = (size_t)NSAMP * NN;      // 16M elems, 64 MB

  // Workspace: Q(4MB) + 4 batched ping-pong buffers (64MB each) = ~260 MB.
  float* Q  = (float*)d_ws;
  float* P0 = Q + NN;
  float* P1 = P0 + BNN;
  float* B0 = P1 + BNN;
  float* B1 = B0 + BNN;

  // 1) Dense OD matrix Q from COO
  zero_f32<<<(unsigned)((NN + 255) / 256), 256, 0, stream>>>(Q, (unsigned)NN);
  scatter_q<<<(NTRIPS + 255) / 256, 256, 0, stream>>>(dsq, trip, Q, NTRIPS);

  // 2) Series init: P = I + L, B = L
  init_PB<<<(unsigned)(BNN / 256), 256, 0, stream>>>(L, P0, B0);

  dim3 grid(NDIM / 128, NDIM / 128, NSAMP);

  // 3) Doubling Neumann product: 6 steps -> sum_{k<=127} L^k
  float *Bc = B0, *Bn = B1, *Pc = P0, *Pn = P1;
  for (int j = 0; j < 6; ++j) {
    gemm128<false, false, false><<<grid, 256, 0, stream>>>(Bc, Bc, nullptr, Bn); // B <- B*B
    gemm128<false, false, true ><<<grid, 256, 0, stream>>>(Pc, Bn, Pc, Pn);      // P <- P + P*B
    float* t;
    t = Bc; Bc = Bn; Bn = t;
    t = Pc; Pc = Pn; Pn = t;
  }
  float* V = Pc;   // V = (I-L)^-1
  float* T = Bc;   // reuse freed B buffers
  float* A = Bn;
  float* W = Pn;   // reuse spare P buffer

  // 4) T = Q .* normalizer(V)
  make_T<<<(unsigned)(BNN / 256), 256, 0, stream>>>(Q, V, T);

  // 5) A = V^T * T ;  W = A * V^T
  gemm128<true,  false, false><<<grid, 256, 0, stream>>>(V, T, nullptr, A);
  gemm128<false, true,  false><<<grid, 256, 0, stream>>>(A, V, nullptr, W);

  // 6) out = L .* W
  final_mul<<<(unsigned)(BNN / 256), 256, 0, stream>>>(L, W, out);
}